// RelativePosOriEncoder_50044958933389
// MI455X (gfx1250) — compile-verified
//
#include <hip/hip_runtime.h>

typedef __attribute__((ext_vector_type(2))) float v2f;
typedef __attribute__((ext_vector_type(8))) float v8f;

#define B_  32
#define G_  512
#define TI  5      // i (and j) values per 16x16 WMMA tile (5*3 = 15 rows/cols used)
#define NTB 103    // ceil(512 / 5)
#define NJG 26     // ceil(103 / 4): j-tile groups, 4 tiles (20 j values) per wave

// ---------------------------------------------------------------------------
// rel_pos[b,i,j,k] = sum_c (p[b,j,c] - p[b,i,c]) * lrf[b,i,c,k]
// One thread: fixed (b,i), 4 consecutive j -> 12 floats = 3 aligned float4 stores.
// ---------------------------------------------------------------------------
__global__ void relpos_kernel(const float* __restrict__ center,
                              const float* __restrict__ lrf,
                              float* __restrict__ out) {
  const int JQ = G_ / 4;
  int tid = blockIdx.x * blockDim.x + threadIdx.x;
  if (tid >= B_ * G_ * JQ) return;
  int jq = tid % JQ;
  int t  = tid / JQ;
  int i  = t % G_;
  int b  = t / G_;
  int j0 = jq * 4;

  const float* pc = center + (size_t)(b * G_ + i) * 6;
  float pix = pc[0], piy = pc[1], piz = pc[2];
  const float* L = lrf + (size_t)(b * G_ + i) * 9;  // L[c*3 + k]
  float L00 = L[0], L01 = L[1], L02 = L[2];
  float L10 = L[3], L11 = L[4], L12 = L[5];
  float L20 = L[6], L21 = L[7], L22 = L[8];

  float r[12];
#pragma unroll
  for (int jj = 0; jj < 4; ++jj) {
    const float* pj = center + (size_t)(b * G_ + j0 + jj) * 6;
    float dx = pj[0] - pix, dy = pj[1] - piy, dz = pj[2] - piz;
    r[jj * 3 + 0] = dx * L00 + dy * L10 + dz * L20;
    r[jj * 3 + 1] = dx * L01 + dy * L11 + dz * L21;
    r[jj * 3 + 2] = dx * L02 + dy * L12 + dz * L22;
  }

  float4* o = (float4*)(out + ((size_t)(b * G_ + i) * G_ + j0) * 3);
  o[0] = make_float4(r[0], r[1], r[2],  r[3]);
  o[1] = make_float4(r[4], r[5], r[6],  r[7]);
  o[2] = make_float4(r[8], r[9], r[10], r[11]);
}

// ---------------------------------------------------------------------------
// rel_ori[b,i,j,m,n] = (L_i^T L_j)[m,n] via V_WMMA_F32_16X16X4_F32.
// One wave: one 5-i block x four 5-j tiles. Per tile:
//   WMMA -> D staged into LDS [row*16+col] (8 ds_store_b32, imm offsets r*64)
//        -> lane p<25 copies pair (il,jl) = (p/5, p%5): 9 contiguous floats to
//           global (compiler-merged wide stores, contiguous 180B per i-row).
// No barriers: each wave touches only its own LDS slice; DS is in-order
// within a wave, fences only pin compiler ordering.
// ---------------------------------------------------------------------------
__global__ void relori_wmma_kernel(const float* __restrict__ lrf,
                                   float* __restrict__ out) {
  __shared__ float tile[8][256];           // 8 waves/block, 16x16 f32 each
  int w    = threadIdx.x >> 5;
  int lane = threadIdx.x & 31;
  int wave = blockIdx.x * (blockDim.x >> 5) + w;

  const int wavesPerB = NTB * NJG;         // 2678
  if (wave >= B_ * wavesPerB) return;      // never fires (grid exact), no barriers used
  int b  = wave / wavesPerB;
  int t  = wave - b * wavesPerB;
  int ib = t / NJG;
  int jg = t - ib * NJG;
  int i0 = ib * TI;

  // ---- operand lane mapping (shared by A and B) ----
  int half = lane >> 4;                    // 0: K=0,1 ; 1: K=2,3
  int M    = lane & 15;
  int l3   = M / 3;                        // i_local (A) / j_local (B)
  int m    = M - l3 * 3;                   // m (A) / n (B)
  int k0   = half ? 2 : 0;

  // ---- A operand: rows (i_local, m) of L_i^T, loaded once per wave ----
  int  i  = i0 + l3;
  bool av = (M < 15) && (i < G_);
  int  ic = (i < G_) ? i : (G_ - 1);
  const float* Li = lrf + (size_t)(b * G_ + ic) * 9;
  v2f a;
  a.x = av ? Li[k0 * 3 + m] : 0.0f;
  a.y = (av && !half) ? Li[3 + m] : 0.0f;  // K=1 (K=3 pad = 0)

  // ---- LDS write base: tile[w][row*16 + col], row = r + half*8, col = M ----
  float* ldsw = &tile[w][half * 128 + M];

  // ---- readback lane mapping: lane -> (il, jl) pair, 25 of 32 lanes ----
  int  ril = lane / 5;
  int  rjl = lane - ril * 5;
  bool rv  = (lane < 25) && (i0 + ril < G_);
  const float* src = &tile[w][ril * 48 + rjl * 3];   // (il*3+m)*16 + jl*3+n
  int j0g = jg * (4 * TI);
  float* dst0 = out + (((size_t)(b * G_ + i0 + ril)) * G_ + j0g + rjl) * 9;

  for (int q = 0; q < 4; ++q) {
    int jb = jg * 4 + q;
    if (jb >= NTB) break;                  // wave-uniform tail check
    int j0q = jb * TI;

    // ---- B operand: cols (j_local, n) of L_j ----
    int  j  = j0q + l3;
    bool bv = (M < 15) && (j < G_);
    int  jc = (j < G_) ? j : (G_ - 1);
    const float* Lj = lrf + (size_t)(b * G_ + jc) * 9;
    v2f bm;
    bm.x = bv ? Lj[k0 * 3 + m] : 0.0f;
    bm.y = (bv && !half) ? Lj[3 + m] : 0.0f;

    v8f c = {};
    v8f d = __builtin_amdgcn_wmma_f32_16x16x4_f32(false, a, false, bm,
                                                  (short)0, c, false, false);

    // ---- stage D into LDS: 8 stores, immediate offsets r*64B ----
    __builtin_amdgcn_fence(__ATOMIC_ACQ_REL, "wavefront");
#pragma unroll
    for (int r = 0; r < 8; ++r) ldsw[r * 16] = d[r];
    __builtin_amdgcn_fence(__ATOMIC_ACQ_REL, "wavefront");

    // ---- coalesced writeback: 9 contiguous floats per valid lane ----
    if (rv && (j0q + rjl) < G_) {
      float* dst = dst0 + q * (TI * 9);
      float v[9];
#pragma unroll
      for (int mm = 0; mm < 3; ++mm)
#pragma unroll
        for (int nn = 0; nn < 3; ++nn)
          v[mm * 3 + nn] = src[mm * 16 + nn];
#pragma unroll
      for (int e = 0; e < 9; ++e) dst[e] = v[e];
    }
    __builtin_amdgcn_fence(__ATOMIC_ACQ_REL, "wavefront");
  }
}

// ---------------------------------------------------------------------------
extern "C" void kernel_launch(void* const* d_in, const int* in_sizes, int n_in,
                              void* d_out, int out_size, void* d_ws, size_t ws_size,
                              hipStream_t stream) {
  const float* center = (const float*)d_in[0];   // (32, 512, 6) f32
  const float* lrf    = (const float*)d_in[1];   // (32, 512, 3, 3) f32
  float* out_pos = (float*)d_out;                               // (32,512,512,3)
  float* out_ori = (float*)d_out + (size_t)B_ * G_ * G_ * 3;    // (32,512,512,3,3)

  // rel_pos: 32*512*128 threads, 256/block
  {
    int threads = B_ * G_ * (G_ / 4);
    int blocks  = (threads + 255) / 256;
    relpos_kernel<<<blocks, 256, 0, stream>>>(center, lrf, out_pos);
  }
  // rel_ori: one wave per (5-i block x 4 j-tiles); 8 waves per 256-thread block
  {
    int totalWaves = B_ * NTB * NJG;             // 32*103*26 = 85,696
    int blocks     = totalWaves / 8;             // 10,712 exactly
    relori_wmma_kernel<<<blocks, 256, 0, stream>>>(lrf, out_ori);
  }
}